// CodebookContrastiveHead6_6743098655124
// MI455X (gfx1250) — compile-verified
//
#include <hip/hip_runtime.h>

typedef float v2f __attribute__((ext_vector_type(2)));
typedef float v8f __attribute__((ext_vector_type(8)));

#define B_TOT  1024
#define Q_TOT  95
#define D_TOT  1024
#define QPC    5
#define NPROTO 6    // prototypes_per_class + 1 (5 positive + 1 background)
#define OUTC   20   // num_classes + 1

// One wave = one (query q, 16-batch tile). C tile is 16x16 f32 (cols 0..5 valid).
__global__ __launch_bounds__(256) void proto_head_kernel(
    const float* __restrict__ qf,   // [B, Q, D]
    const float* __restrict__ wt,   // [114, D]
    float* __restrict__ out)        // [B, Q, 20]
{
    __shared__ float lds[8][16 * 17];   // per-wave 16x16 tile, padded stride

    const int lane = threadIdx.x & 31;
    const int w    = threadIdx.x >> 5;            // wave in block: 0..7
    const int q    = blockIdx.x;                  // 0..94
    const int b0   = (blockIdx.y * 8 + w) * 16;   // batch tile base
    const int cls  = q / QPC;
    const int half = lane >> 4;                   // 0: lanes 0-15, 1: lanes 16-31
    const int mn   = lane & 15;                   // M for A / N for B
    const int klo  = half * 2;                    // K offset within 4-step

    // A: row m = mn of the 16-batch tile for query q; lane loads K = {klo, klo+1}
    const float* Arow = qf + ((size_t)(b0 + mn) * Q_TOT + q) * D_TOT + klo;
    // B: column n = mn -> weight row cls*6 + n (clamped; cols >=6 are discarded)
    const int wr = cls * NPROTO + (mn < NPROTO ? mn : NPROTO - 1);
    const float* Brow = wt + (size_t)wr * D_TOT + klo;

    v8f acc0 = {};
    v8f acc1 = {};
    for (int k = 0; k < D_TOT; k += 32) {
        // One speculative prefetch per 128B of streamed A data, ~2KB ahead.
        // OOB speculative prefetches are silently dropped (no guard needed).
        __builtin_prefetch(Arow + k + 512, 0, 3);   // global_prefetch_b8
        #pragma unroll
        for (int kk = 0; kk < 32; kk += 8) {
            // A is streamed once -> non-temporal; B is hot in WGP$/L2 -> regular
            v2f a0 = __builtin_nontemporal_load((const v2f*)(Arow + k + kk));
            v2f bb0 = *(const v2f*)(Brow + k + kk);
            v2f a1 = __builtin_nontemporal_load((const v2f*)(Arow + k + kk + 4));
            v2f bb1 = *(const v2f*)(Brow + k + kk + 4);
            // (neg_a, A, neg_b, B, c_mod, C, reuse_a, reuse_b)
            acc0 = __builtin_amdgcn_wmma_f32_16x16x4_f32(
                false, a0, false, bb0, (short)0, acc0, false, false);
            acc1 = __builtin_amdgcn_wmma_f32_16x16x4_f32(
                false, a1, false, bb1, (short)0, acc1, false, false);
        }
    }
    v8f acc = acc0 + acc1;

    // C/D layout: VGPR r -> (M = half*8 + r, N = mn). Spill tile to LDS.
    #pragma unroll
    for (int r = 0; r < 8; ++r)
        lds[w][(half * 8 + r) * 17 + mn] = acc[r];

    __syncthreads();

    // Lanes 0..15: reduce one row each (max of 5 positives, background col 5),
    // then write the 20-wide -inf-filled logits row.
    if (lane < 16) {
        const float* rowp = &lds[w][lane * 17];
        float pm = rowp[0];
        pm = fmaxf(pm, rowp[1]);
        pm = fmaxf(pm, rowp[2]);
        pm = fmaxf(pm, rowp[3]);
        pm = fmaxf(pm, rowp[4]);
        const float bg = rowp[5];
        float* o = out + ((size_t)(b0 + lane) * Q_TOT + q) * OUTC;
        #pragma unroll
        for (int j = 0; j < OUTC; ++j) {
            float v = -__builtin_huge_valf();
            if (j == cls)      v = pm;
            if (j == OUTC - 1) v = bg;
            o[j] = v;
        }
    }
}

extern "C" void kernel_launch(void* const* d_in, const int* in_sizes, int n_in,
                              void* d_out, int out_size, void* d_ws, size_t ws_size,
                              hipStream_t stream) {
    (void)in_sizes; (void)n_in; (void)out_size; (void)d_ws; (void)ws_size;
    const float* qf = (const float*)d_in[0];   // [1024, 95, 1024] f32
    const float* wt = (const float*)d_in[1];   // [114, 1024] f32
    float* out = (float*)d_out;                // [1024, 95, 20] f32

    dim3 grid(Q_TOT, B_TOT / (16 * 8));        // 95 x 8 blocks
    proto_head_kernel<<<grid, 256, 0, stream>>>(qf, wt, out);
}